// DyGMixer_34961033790096
// MI455X (gfx1250) — compile-verified
//
#include <hip/hip_runtime.h>
#include <hip/hip_bf16.h>

// Problem constants from the reference: B=256, L=512, D=64, ids in [0,2000)
#define LSEQ   512
#define DFEAT  64
#define NBINS  2048          // >= 2000, power of two
#define TROWS  544           // 34 * 16 >= 513 possible count values (0..512)

typedef __attribute__((ext_vector_type(2))) float v2f;
typedef __attribute__((ext_vector_type(8))) float v8f;

// ---------------------------------------------------------------------------
// Kernel 1: build lookup table T[v][e] = sum_d relu(v*w1[d]+b1[d]) * W2[e][d] + b2[e]
// One wave (32 threads) per block; each block computes 16 rows x 64 cols via
// V_WMMA_F32_16X16X4_F32 (4 N-tiles x 16 K-steps, K=4 each).
// ---------------------------------------------------------------------------
__global__ __launch_bounds__(32) void dyg_table_wmma(
    const float* __restrict__ w1, const float* __restrict__ b1,
    const float* __restrict__ W2, const float* __restrict__ b2,
    float* __restrict__ table)
{
    const int lane = threadIdx.x;        // 0..31
    const int half = lane >> 4;          // 0 or 1 (selects K upper pair / row upper half)
    const int l15  = lane & 15;
    const int v0   = blockIdx.x * 16;    // first count value of this tile
    const float v  = (float)(v0 + l15);  // A-matrix row index m = lane&15 (both halves)

    #pragma unroll
    for (int nt = 0; nt < 4; ++nt) {
        const int e0 = nt * 16;
        const int e  = e0 + l15;         // B/C/D column n = lane&15

        // Seed accumulator with b2[n] once per D element -> bias folded in.
        v8f c;
        const float bb = b2[e];
        #pragma unroll
        for (int r = 0; r < 8; ++r) c[r] = bb;

        #pragma unroll 4
        for (int kb = 0; kb < 16; ++kb) {
            const int d0 = kb * 4 + half * 2;   // K = vgpr + 2*half within the 4-wide step
            v2f a, bmat;
            // A[m][k] = relu(v*w1[k] + b1[k])
            a.x = fmaxf(v * w1[d0]     + b1[d0],     0.0f);
            a.y = fmaxf(v * w1[d0 + 1] + b1[d0 + 1], 0.0f);
            // B[k][n] = W2[n][k]  (einsum 'blcd,ed->blce' contracts W2 over its 2nd dim)
            bmat.x = W2[e * DFEAT + d0];
            bmat.y = W2[e * DFEAT + d0 + 1];
            // 8 args: (neg_a, A, neg_b, B, c_mod, C, reuse_a, reuse_b)
            c = __builtin_amdgcn_wmma_f32_16x16x4_f32(
                    false, a, false, bmat, (short)0, c, false, false);
        }

        // D layout: VGPR r, lane -> row m = r + 8*half, col n = lane&15
        #pragma unroll
        for (int r = 0; r < 8; ++r) {
            const int vv = v0 + r + 8 * half;
            table[vv * DFEAT + e] = c[r];
        }
    }
}

// ---------------------------------------------------------------------------
// Kernel 2: per-batch LDS histograms + table gather-add + streaming stores.
// One block (256 threads = 8 waves) per batch row. Output-write bound.
// ---------------------------------------------------------------------------
__global__ __launch_bounds__(256) void dyg_hist_gather(
    const int* __restrict__ src_ids, const int* __restrict__ dst_ids,
    const float* __restrict__ table,
    float* __restrict__ out_src, float* __restrict__ out_dst)
{
    __shared__ int hist_s[NBINS];
    __shared__ int hist_d[NBINS];

    const int b = blockIdx.x;
    const int t = threadIdx.x;

    for (int i = t; i < NBINS; i += 256) { hist_s[i] = 0; hist_d[i] = 0; }
    __syncthreads();

    const int* __restrict__ srow = src_ids + (size_t)b * LSEQ;
    const int* __restrict__ drow = dst_ids + (size_t)b * LSEQ;

    for (int i = t; i < LSEQ; i += 256) {
        atomicAdd(&hist_s[srow[i] & (NBINS - 1)], 1);
        atomicAdd(&hist_d[drow[i] & (NBINS - 1)], 1);
    }
    __syncthreads();

    // 16 groups of 16 threads; group handles positions i = grp, grp+16, ...
    // thread-in-group handles one float4 chunk of the 64-float output row.
    const int grp   = t >> 4;
    const int chunk = t & 15;

    #pragma unroll
    for (int arr = 0; arr < 2; ++arr) {
        const int* __restrict__ row = arr ? drow : srow;
        float* __restrict__ out = (arr ? out_dst : out_src) + (size_t)b * LSEQ * DFEAT;

        for (int i = grp; i < LSEQ; i += 16) {
            const int id = row[i];
            const int c0 = (id == 0) ? 0 : hist_s[id & (NBINS - 1)];
            const int c1 = (id == 0) ? 0 : hist_d[id & (NBINS - 1)];
            const float4 a  = ((const float4*)(table + (size_t)c0 * DFEAT))[chunk];
            const float4 bb = ((const float4*)(table + (size_t)c1 * DFEAT))[chunk];
            float4 r;
            r.x = a.x + bb.x; r.y = a.y + bb.y; r.z = a.z + bb.z; r.w = a.w + bb.w;
            ((float4*)(out + (size_t)i * DFEAT))[chunk] = r;
        }
    }
}

extern "C" void kernel_launch(void* const* d_in, const int* in_sizes, int n_in,
                              void* d_out, int out_size, void* d_ws, size_t ws_size,
                              hipStream_t stream) {
    const int*   src_ids = (const int*)d_in[0];
    const int*   dst_ids = (const int*)d_in[1];
    const float* w1      = (const float*)d_in[2];
    const float* b1      = (const float*)d_in[3];
    const float* W2      = (const float*)d_in[4];
    const float* b2      = (const float*)d_in[5];

    const int B = in_sizes[0] / LSEQ;          // 256
    float* table   = (float*)d_ws;             // TROWS x 64 f32 = 139 KB scratch
    float* out_src = (float*)d_out;
    float* out_dst = out_src + (size_t)B * LSEQ * DFEAT;

    dyg_table_wmma<<<TROWS / 16, 32, 0, stream>>>(w1, b1, W2, b2, table);
    dyg_hist_gather<<<B, 256, 0, stream>>>(src_ids, dst_ids, table, out_src, out_dst);
}